// Voxelization_62749472195254
// MI455X (gfx1250) — compile-verified
//
#include <hip/hip_runtime.h>
#include <hip/hip_bf16.h>

// Voxelization for MI455X (gfx1250, wave32).
// Pipeline:
//   1) mean_kernel    : per (b,dim) mean over N, WMMA-f32 ones-matrix reduction
//   2) maxnorm_kernel : per-batch max squared norm -> sqrt (float4 loads)
//   3) voxelize_kernel: write nc (output #2) + flat voxel index (vectorized)
//   4) counts_kernel  : per-batch voxel counts via LDS atomics (128KB LDS grid)
//   5) accum_kernel   : per (b,c) LDS-resident 32^3 grid, ds_add_f32 scatter,
//                       b128 loads/stores, then divide by max(count,1)

typedef __attribute__((ext_vector_type(2))) float v2f;
typedef __attribute__((ext_vector_type(8))) float v8f;

#define RV 32
#define R3 (RV * RV * RV)   // 32768
#define NB 8
#define NC 64
#define NPT 65536

// ---------------------------------------------------------------- kernel 1
// One block per (b*3+d) row; 8 waves, each wave reduces an 8192-float chunk
// with V_WMMA_F32_16X16X4_F32 (A = ones), 64 floats per WMMA.
__global__ void vox_mean_kernel(const float* __restrict__ coords,
                                float* __restrict__ means, int N) {
  const int row  = blockIdx.x;            // 0..23
  const int wave = threadIdx.x >> 5;      // 0..7
  const int lane = threadIdx.x & 31;
  const float* __restrict__ src = coords + (size_t)row * N;
  const int chunk = N >> 3;               // 8192
  const int base  = wave * chunk;

  v2f ones = {1.0f, 1.0f};
  v8f c = {};
  for (int i = 0; i < chunk; i += 64) {
    v2f bm;
    bm.x = src[base + i + lane];
    bm.y = src[base + i + 32 + lane];
    // D[i][j] = sum_k B[k][j] + C[i][j]  (A all-ones)
    c = __builtin_amdgcn_wmma_f32_16x16x4_f32(
        /*neg_a=*/false, ones, /*neg_b=*/false, bm,
        /*c_mod=*/(short)0, c, /*reuse_a=*/false, /*reuse_b=*/false);
  }
  // c[0]: lanes 0-15 = D[0][j], lanes 16-31 = D[8][j] (identical rows)
  // -> sum over all 32 lanes equals 2 * (total sum of this wave's chunk).
  float partial = c[0];
  for (int off = 16; off >= 1; off >>= 1)
    partial += __shfl_xor(partial, off, 32);

  __shared__ float wsum[8];
  if (lane == 0) wsum[wave] = partial * 0.5f;
  __syncthreads();
  if (threadIdx.x == 0) {
    float t = 0.0f;
    for (int w = 0; w < 8; ++w) t += wsum[w];
    means[row] = t / (float)N;
  }
}

// ---------------------------------------------------------------- kernel 2
__global__ void vox_maxnorm_kernel(const float* __restrict__ coords,
                                   const float* __restrict__ means,
                                   float* __restrict__ maxn, int N) {
  const int b = blockIdx.x;
  const int K = N >> 2;   // float4 count per row
  const float4* __restrict__ cx = (const float4*)(coords + (size_t)b * 3 * N);
  const float4* __restrict__ cy = (const float4*)(coords + (size_t)b * 3 * N + N);
  const float4* __restrict__ cz = (const float4*)(coords + (size_t)b * 3 * N + 2 * N);
  const float mx = means[b * 3 + 0];
  const float my = means[b * 3 + 1];
  const float mz = means[b * 3 + 2];
  float m = 0.0f;
  for (int i = threadIdx.x; i < K; i += blockDim.x) {
    float4 x = cx[i], y = cy[i], z = cz[i];
    float dx, dy, dz;
    dx = x.x - mx; dy = y.x - my; dz = z.x - mz; m = fmaxf(m, dx*dx + dy*dy + dz*dz);
    dx = x.y - mx; dy = y.y - my; dz = z.y - mz; m = fmaxf(m, dx*dx + dy*dy + dz*dz);
    dx = x.z - mx; dy = y.z - my; dz = z.z - mz; m = fmaxf(m, dx*dx + dy*dy + dz*dz);
    dx = x.w - mx; dy = y.w - my; dz = z.w - mz; m = fmaxf(m, dx*dx + dy*dy + dz*dz);
  }
  for (int off = 16; off >= 1; off >>= 1)
    m = fmaxf(m, __shfl_xor(m, off, 32));
  __shared__ float wm[8];
  if ((threadIdx.x & 31) == 0) wm[threadIdx.x >> 5] = m;
  __syncthreads();
  if (threadIdx.x == 0) {
    float t = wm[0];
    for (int w = 1; w < 8; ++w) t = fmaxf(t, wm[w]);
    maxn[b] = sqrtf(t);
  }
}

// ---------------------------------------------------------------- kernel 3
// 4 points per thread: float4 coord loads, float4 nc stores, int4 vox stores.
__global__ void vox_voxelize_kernel(const float* __restrict__ coords,
                                    const float* __restrict__ means,
                                    const float* __restrict__ maxn,
                                    float* __restrict__ nc_out,
                                    int* __restrict__ voxflat, int N, int B) {
  const int K = N >> 2;                        // float4 per (b,dim) row
  const int idx = blockIdx.x * blockDim.x + threadIdx.x;
  if (idx >= B * K) return;
  const int b  = idx / K;
  const int n4 = idx - b * K;
  const float inv = 1.0f / (2.0f * maxn[b]);   // EPS == 0

  int4 flat = {0, 0, 0, 0};
#pragma unroll
  for (int d = 0; d < 3; ++d) {
    const float mu = means[b * 3 + d];
    float4 v = ((const float4*)(coords + ((size_t)b * 3 + d) * N))[n4];
    float4 t;
    t.x = fminf(fmaxf(((v.x - mu) * inv + 0.5f) * (float)RV, 0.0f), (float)(RV - 1));
    t.y = fminf(fmaxf(((v.y - mu) * inv + 0.5f) * (float)RV, 0.0f), (float)(RV - 1));
    t.z = fminf(fmaxf(((v.z - mu) * inv + 0.5f) * (float)RV, 0.0f), (float)(RV - 1));
    t.w = fminf(fmaxf(((v.w - mu) * inv + 0.5f) * (float)RV, 0.0f), (float)(RV - 1));
    ((float4*)(nc_out + ((size_t)b * 3 + d) * N))[n4] = t;
    flat.x = flat.x * RV + (int)rintf(t.x);    // round-half-even == jnp.round
    flat.y = flat.y * RV + (int)rintf(t.y);
    flat.z = flat.z * RV + (int)rintf(t.z);
    flat.w = flat.w * RV + (int)rintf(t.w);
  }
  ((int4*)(voxflat + (size_t)b * N))[n4] = flat;
}

// ---------------------------------------------------------------- kernel 4
// One block per batch; 128KB dynamic LDS of uint counts; int4 index loads.
__global__ void vox_counts_kernel(const int* __restrict__ voxflat,
                                  float* __restrict__ counts, int N) {
  extern __shared__ unsigned int cnt[];
  const int b = blockIdx.x;
  for (int i = threadIdx.x; i < R3; i += blockDim.x) cnt[i] = 0u;
  __syncthreads();
  const int4* __restrict__ vf = (const int4*)(voxflat + (size_t)b * N);
  const int K = N >> 2;
  for (int i = threadIdx.x; i < K; i += blockDim.x) {
    int4 v = vf[i];
    atomicAdd(&cnt[v.x], 1u);
    atomicAdd(&cnt[v.y], 1u);
    atomicAdd(&cnt[v.z], 1u);
    atomicAdd(&cnt[v.w], 1u);
  }
  __syncthreads();
  float4* __restrict__ co = (float4*)(counts + (size_t)b * R3);
  for (int i = threadIdx.x; i < (R3 >> 2); i += blockDim.x) {
    float4 c;
    c.x = (float)cnt[4 * i + 0];
    c.y = (float)cnt[4 * i + 1];
    c.z = (float)cnt[4 * i + 2];
    c.w = (float)cnt[4 * i + 3];
    co[i] = c;
  }
}

// ---------------------------------------------------------------- kernel 5
// One block per (b,c); 128KB dynamic LDS float grid; b128 global traffic,
// ds_add_f32 scatter, fused average on write-out.
__global__ void vox_accum_kernel(const float* __restrict__ features,
                                 const int* __restrict__ voxflat,
                                 const float* __restrict__ counts,
                                 float* __restrict__ out, int N) {
  extern __shared__ float acc[];
  const int g = blockIdx.x;          // b*NC + c
  const int b = g >> 6;              // NC == 64
  for (int i = threadIdx.x; i < R3; i += blockDim.x) acc[i] = 0.0f;
  __syncthreads();
  const float4* __restrict__ f  = (const float4*)(features + (size_t)g * N);
  const int4*   __restrict__ vf = (const int4*)(voxflat + (size_t)b * N);
  const int K = N >> 2;
  for (int i = threadIdx.x; i < K; i += blockDim.x) {
    __builtin_prefetch(f + i + 2 * blockDim.x, 0, 0);   // global_prefetch_b8
    float4 fv = f[i];                                    // global_load_b128
    int4   v  = vf[i];
    atomicAdd(&acc[v.x], fv.x);                          // ds_add_f32 x4
    atomicAdd(&acc[v.y], fv.y);
    atomicAdd(&acc[v.z], fv.z);
    atomicAdd(&acc[v.w], fv.w);
  }
  __syncthreads();
  float4* __restrict__ o = (float4*)(out + (size_t)g * R3);
  const float4* __restrict__ cb = (const float4*)(counts + (size_t)b * R3);
  for (int i = threadIdx.x; i < (R3 >> 2); i += blockDim.x) {
    float4 s;
    s.x = acc[4 * i + 0];
    s.y = acc[4 * i + 1];
    s.z = acc[4 * i + 2];
    s.w = acc[4 * i + 3];
    float4 c = cb[i];
    float4 r;
    r.x = s.x / fmaxf(c.x, 1.0f);
    r.y = s.y / fmaxf(c.y, 1.0f);
    r.z = s.z / fmaxf(c.z, 1.0f);
    r.w = s.w / fmaxf(c.w, 1.0f);
    o[i] = r;                                            // global_store_b128
  }
}

// ---------------------------------------------------------------- launcher
extern "C" void kernel_launch(void* const* d_in, const int* in_sizes, int n_in,
                              void* d_out, int out_size, void* d_ws, size_t ws_size,
                              hipStream_t stream) {
  const float* features = (const float*)d_in[0];   // [8,64,65536]
  const float* coords   = (const float*)d_in[1];   // [8,3,65536]

  const int B = NB, N = NPT;

  // Output: [ out (B*C*R^3) | nc (B*3*N) ]
  float* out_vox = (float*)d_out;
  float* out_nc  = out_vox + (size_t)NB * NC * R3;

  // Workspace layout (16B-aligned sections)
  float* means   = (float*)d_ws;                      // 24 floats
  float* maxn    = means + 32;                        // 8 floats
  int*   voxflat = (int*)(maxn + 32);                 // B*N ints (2MB), 256B off
  float* counts  = (float*)(voxflat + (size_t)B * N); // B*R^3 floats (1MB)

  vox_mean_kernel<<<B * 3, 256, 0, stream>>>(coords, means, N);
  vox_maxnorm_kernel<<<B, 256, 0, stream>>>(coords, means, maxn, N);
  vox_voxelize_kernel<<<(B * (N / 4) + 255) / 256, 256, 0, stream>>>(
      coords, means, maxn, out_nc, voxflat, N, B);
  vox_counts_kernel<<<B, 256, R3 * sizeof(unsigned int), stream>>>(
      voxflat, counts, N);
  vox_accum_kernel<<<B * NC, 256, R3 * sizeof(float), stream>>>(
      features, voxflat, counts, out_vox, N);
}